// CTCLOSS_60344290508922
// MI455X (gfx1250) — compile-verified
//
#include <hip/hip_runtime.h>
#include <stdint.h>

// Problem constants (from reference)
#define Bv 64
#define Tv 512
#define Vv 1024
#define Sv 128
#define Lv 257            // 2*S + 1
#define NEGF (-1.0e30f)

// ---------------------------------------------------------------------------
// Kernel 1: per-(b,t) logsumexp over V=1024 + gather of the 257 extended-label
// log-probs into glp[b][t][s] = pred[b,t,ext[s]] - lse(b,t).
// One wave per row; the 4KB row is staged into LDS with CDNA5 async
// global->LDS b128 loads, reductions via wave32 shuffles.
// ---------------------------------------------------------------------------
__global__ __launch_bounds__(256) void lse_gather_kernel(
    const float* __restrict__ pred,     // [B,T,V]
    const int*   __restrict__ targets,  // [B,S]
    const int*   __restrict__ plen,     // [B]
    float*       __restrict__ glp)      // [B,T,L] workspace
{
    __shared__ __align__(16) float smem[8 * Vv];   // 8 waves * 4KB
    const int wave = threadIdx.x >> 5;
    const int lane = threadIdx.x & 31;
    const int row  = blockIdx.x * 8 + wave;        // row = b*T + t
    const int b    = row >> 9;                     // /T (T=512)
    const int t    = row & (Tv - 1);

    if (t >= plen[b]) return;  // alpha never reads rows with t >= in_len (wave-uniform)

    // ---- async stage: 4KB row -> LDS slot (8 x b128 per lane) ----
    const unsigned long long gbase = (unsigned long long)(uintptr_t)pred;
    const unsigned gofs = (unsigned)row * (Vv * 4u) + (unsigned)lane * 16u;
    const unsigned lofs = (unsigned)(uintptr_t)(&smem[wave * Vv]) + (unsigned)lane * 16u;
#pragma unroll
    for (int j = 0; j < 8; ++j) {
        asm volatile("global_load_async_to_lds_b128 %0, %1, %2"
                     :: "v"(lofs + 512u * j), "v"(gofs + 512u * j), "s"(gbase)
                     : "memory");
    }
    asm volatile("s_wait_asynccnt 0x0" ::: "memory");

    // ---- read back (conflict-free ds_load_b128 pattern) + reduce ----
    const float* slot = &smem[wave * Vv];
    float4 vv[8];
#pragma unroll
    for (int j = 0; j < 8; ++j)
        vv[j] = *(const float4*)&slot[j * 128 + lane * 4];

    float mx = -3.4e38f;
#pragma unroll
    for (int j = 0; j < 8; ++j)
        mx = fmaxf(mx, fmaxf(fmaxf(vv[j].x, vv[j].y), fmaxf(vv[j].z, vv[j].w)));
#pragma unroll
    for (int off = 16; off > 0; off >>= 1)
        mx = fmaxf(mx, __shfl_xor(mx, off));

    float se = 0.f;
#pragma unroll
    for (int j = 0; j < 8; ++j) {
        se += expf(vv[j].x - mx);
        se += expf(vv[j].y - mx);
        se += expf(vv[j].z - mx);
        se += expf(vv[j].w - mx);
    }
#pragma unroll
    for (int off = 16; off > 0; off >>= 1)
        se += __shfl_xor(se, off);

    const float lse = mx + logf(se);

    // ---- gather the 257 extended labels from the LDS-resident row ----
    const int* tg = targets + b * Sv;
    float* outRow = glp + (long)row * Lv;
    for (int s = lane; s < Lv; s += 32) {
        const int lab = (s & 1) ? tg[s >> 1] : 0;   // even -> blank(0)
        outRow[s] = slot[lab] - lse;
    }
}

// ---------------------------------------------------------------------------
// Kernel 2: CTC alpha recursion. One wave32 per batch; 9 states per lane in
// registers; neighbor states via shfl_up; glp row for t+1 prefetched while
// computing step t. Loop runs only t = 1..in_len-1 (alpha frozen after).
// ---------------------------------------------------------------------------
__global__ __launch_bounds__(32) void ctc_alpha_kernel(
    const float* __restrict__ glp,      // [B,T,L]
    const int*   __restrict__ targets,  // [B,S]
    const int*   __restrict__ plen,     // [B]
    const int*   __restrict__ tlen,     // [B]
    float*       __restrict__ losses)   // [B]
{
    const int b    = blockIdx.x;
    const int lane = threadIdx.x;
    const int s0   = lane * 9;
    const int* tg  = targets + b * Sv;

    // skip_ok[s]: s odd, s>=3, tgt[s/2] != tgt[s/2-1]  (even s -> blank -> false)
    bool skipok[9];
#pragma unroll
    for (int k = 0; k < 9; ++k) {
        const int s = s0 + k;
        bool ok = false;
        if (s < Lv && (s & 1) && s >= 3)
            ok = (tg[s >> 1] != tg[(s >> 1) - 1]);
        skipok[k] = ok;
    }

    const float* gb = glp + (long)b * Tv * Lv;

    float A[9];
#pragma unroll
    for (int k = 0; k < 9; ++k) {
        const int s = s0 + k;
        A[k] = (s <= 1) ? gb[s] : NEGF;    // alpha[0]=lp[0,0], alpha[1]=lp[0,1]
    }

    const int inLen = plen[b];

    float g[9], gn[9];
#pragma unroll
    for (int k = 0; k < 9; ++k) {          // preload row t=1
        const int s = s0 + k;
        g[k] = (s < Lv) ? gb[(long)Lv + s] : 0.f;
    }

    for (int t = 1; t < inLen; ++t) {
        // prefetch next step's row (clamped; unused values harmless)
        const int tn = (t + 1 < Tv) ? (t + 1) : t;
        const float* rn = gb + (long)tn * Lv;
#pragma unroll
        for (int k = 0; k < 9; ++k) {
            const int s = s0 + k;
            gn[k] = (s < Lv) ? rn[s] : 0.f;
        }

        const float pA8 = __shfl_up(A[8], 1);   // prev lane's state s0-1
        const float pA7 = __shfl_up(A[7], 1);   // prev lane's state s0-2

        float Anew[9];
#pragma unroll
        for (int k = 0; k < 9; ++k) {
            const int s = s0 + k;
            const float a0 = A[k];
            float a1 = (k > 0) ? A[k - 1] : pA8;
            if (s == 0) a1 = NEGF;
            float a2 = (k > 1) ? A[k - 2] : ((k == 1) ? pA8 : pA7);
            if (!skipok[k]) a2 = NEGF;

            const float m    = fmaxf(a0, fmaxf(a1, a2));
            const float comb = m + logf(expf(a0 - m) + expf(a1 - m) + expf(a2 - m));
            Anew[k] = comb + g[k];
        }
#pragma unroll
        for (int k = 0; k < 9; ++k) { A[k] = Anew[k]; g[k] = gn[k]; }
    }

    // final: loss = -logaddexp(alpha[2*tl], alpha[2*tl-1]); clamp "no path" to 0
    __shared__ float sA[Lv];
#pragma unroll
    for (int k = 0; k < 9; ++k) {
        const int s = s0 + k;
        if (s < Lv) sA[s] = A[k];
    }
    __syncthreads();
    if (lane == 0) {
        const int e = 2 * tlen[b];
        const float x = sA[e], y = sA[e - 1];
        const float m = fmaxf(x, y);
        const float ll = m + log1pf(expf(fminf(x, y) - m));
        float loss = -ll;
        if (loss > 1e29f) loss = 0.f;
        losses[b] = loss;
    }
}

// ---------------------------------------------------------------------------
// Kernel 3: deterministic fixed-order reduction of the 64 per-batch losses.
// ---------------------------------------------------------------------------
__global__ __launch_bounds__(32) void reduce_losses_kernel(
    const float* __restrict__ losses, float* __restrict__ out)
{
    const int lane = threadIdx.x;
    float v = losses[lane] + losses[lane + 32];
#pragma unroll
    for (int off = 16; off > 0; off >>= 1)
        v += __shfl_xor(v, off);
    if (lane == 0) out[0] = v;
}

extern "C" void kernel_launch(void* const* d_in, const int* in_sizes, int n_in,
                              void* d_out, int out_size, void* d_ws, size_t ws_size,
                              hipStream_t stream) {
    (void)in_sizes; (void)n_in; (void)out_size; (void)ws_size;
    const float* pred    = (const float*)d_in[0];
    const int*   targets = (const int*)d_in[1];
    const int*   plen    = (const int*)d_in[2];
    const int*   tlen    = (const int*)d_in[3];

    float* glp    = (float*)d_ws;                                        // B*T*L floats (33.7 MB)
    float* losses = (float*)((char*)d_ws + (size_t)Bv * Tv * Lv * sizeof(float));

    lse_gather_kernel<<<dim3((Bv * Tv) / 8), dim3(256), 0, stream>>>(pred, targets, plen, glp);
    ctc_alpha_kernel<<<dim3(Bv), dim3(32), 0, stream>>>(glp, targets, plen, tlen, losses);
    reduce_losses_kernel<<<dim3(1), dim3(32), 0, stream>>>(losses, (float*)d_out);
}